// Conv2D_6674379178131
// MI455X (gfx1250) — compile-verified
//
#include <hip/hip_runtime.h>
#include <cstdint>

// ---------------------------------------------------------------------------
// 3x3 VALID conv, 4096x4096 fp32 -> 4094x4094 fp32, + bias.
// Memory-bound (AI ~2.25 flop/byte): optimize for HBM streaming.
// CDNA5 path: Tensor Data Mover DMAs each 66x66 halo tile into LDS
// (TENSORcnt-tracked), stencil computed from LDS with register rolling.
// ---------------------------------------------------------------------------

#define IN_W   4096
#define OUT_W  4094
#define TILE   64
#define HALO   66            // TILE + 2
#define LDS_FLOATS (HALO * HALO)

typedef __attribute__((ext_vector_type(4))) unsigned int v4u;
typedef __attribute__((ext_vector_type(8))) int          v8i;
typedef __attribute__((ext_vector_type(4))) int          v4i;

#if defined(__has_builtin)
#if __has_builtin(__builtin_amdgcn_tensor_load_to_lds) && __has_builtin(__builtin_amdgcn_s_wait_tensorcnt)
#define USE_TDM 1
#endif
#endif
#ifndef USE_TDM
#define USE_TDM 0
#endif

__global__ __launch_bounds__(256) void conv3x3_tdm_kernel(
    const float* __restrict__ in,
    const float* __restrict__ wgt,
    const float* __restrict__ bias,
    float* __restrict__ out)
{
    extern __shared__ float tile[];   // HALO*HALO floats, at dynamic-LDS offset 0

    const int col0 = blockIdx.x * TILE;
    const int row0 = blockIdx.y * TILE;
    const int tid  = threadIdx.x;

#if USE_TDM
    // ---- Wave 0 issues one TDM 2D tile load: global (66x66 window) -> LDS ----
    if (tid < 32) {
        const unsigned long long gaddr =
            (unsigned long long)(uintptr_t)in +
            ((unsigned long long)(row0 * IN_W + col0)) * 4ull;
        // Remaining tensor extent measured from the tile-start address:
        // elements past it read back as zero (right/bottom edge handling).
        const unsigned rem0 = (unsigned)(IN_W - col0);   // elems per line remaining
        const unsigned rem1 = (unsigned)(IN_W - row0);   // lines remaining

        v4u g0;
        g0.x = 1u;                                        // count=1 (valid), user mode
        g0.y = 0u;                                        // lds_addr = 0 (dyn LDS base)
        g0.z = (unsigned)(gaddr & 0xFFFFFFFFull);         // global_addr[31:0]
        g0.w = (unsigned)((gaddr >> 32) & 0x01FFFFFFull)  // global_addr[56:32]
             | 0x80000000u;                               // type = 2 ("image")

        v8i g1;
        g1[0] = 0x00020000;                               // wg_mask=0, data_size=2 (4B)
        g1[1] = (int)((rem0 & 0xFFFFu) << 16);            // tensor_dim0[15:0]
        g1[2] = (int)((rem0 >> 16) | ((rem1 & 0xFFFFu) << 16)); // dim0[31:16]|dim1[15:0]
        g1[3] = (int)((rem1 >> 16) | (HALO << 16));       // dim1[31:16] | tile_dim0=66
        g1[4] = HALO;                                     // tile_dim1=66, tile_dim2=0
        g1[5] = IN_W;                                     // tensor_dim0_stride[31:0]
        g1[6] = 0;                                        // stride0[47:32], stride1[15:0]
        g1[7] = 0;                                        // stride1[47:16]

        v4i g2 = {0, 0, 0, 0};                            // 2D tensor: groups 2/3 unused
        v4i g3 = {0, 0, 0, 0};
        v8i g4 = {0, 0, 0, 0, 0, 0, 0, 0};                // extra group (clang-23 form)

        __builtin_amdgcn_tensor_load_to_lds(g0, g1, g2, g3, g4, 0);
        __builtin_amdgcn_s_wait_tensorcnt(0);             // TENSORcnt == 0
    }
#else
    // Fallback: cooperative global -> LDS copy (no TDM builtin available).
    for (int i = tid; i < LDS_FLOATS; i += 256) {
        const int r  = i / HALO;
        const int c  = i - r * HALO;
        const int gr = row0 + r;
        const int gc = col0 + c;
        tile[i] = (gr < IN_W && gc < IN_W) ? in[(size_t)gr * IN_W + gc] : 0.0f;
    }
#endif
    __syncthreads();

    // ---- Stencil from LDS: 256 threads = 64 cols x 4 row-groups of 16 rows ----
    const int tx = tid & 63;          // column within tile (lane-contiguous)
    const int ty = tid >> 6;          // row-group 0..3
    const int ocol = col0 + tx;

    const float w00 = wgt[0], w01 = wgt[1], w02 = wgt[2];
    const float w10 = wgt[3], w11 = wgt[4], w12 = wgt[5];
    const float w20 = wgt[6], w21 = wgt[7], w22 = wgt[8];
    const float b = bias[0];

    if (ocol >= OUT_W) return;        // no barriers after this point

    const int rbase = ty * 16;        // 16 consecutive rows -> vertical reuse
    const float* t0 = tile + rbase * HALO + tx;

    // Rolling 3x3 register window: each LDS row loaded once, used 3x.
    float c00 = t0[0],        c01 = t0[1],        c02 = t0[2];
    float c10 = t0[HALO + 0], c11 = t0[HALO + 1], c12 = t0[HALO + 2];

#pragma unroll
    for (int i = 0; i < 16; ++i) {
        const float* t2 = tile + (rbase + i + 2) * HALO + tx;
        const float c20 = t2[0], c21 = t2[1], c22 = t2[2];

        float acc = fmaf(w00, c00,
                    fmaf(w01, c01,
                    fmaf(w02, c02,
                    fmaf(w10, c10,
                    fmaf(w11, c11,
                    fmaf(w12, c12,
                    fmaf(w20, c20,
                    fmaf(w21, c21,
                    fmaf(w22, c22, b)))))))));

        const int orow = row0 + rbase + i;
        if (orow < OUT_W)
            out[(size_t)orow * OUT_W + ocol] = acc;

        c00 = c10; c01 = c11; c02 = c12;
        c10 = c20; c11 = c21; c12 = c22;
    }
}

extern "C" void kernel_launch(void* const* d_in, const int* in_sizes, int n_in,
                              void* d_out, int out_size, void* d_ws, size_t ws_size,
                              hipStream_t stream) {
    (void)in_sizes; (void)n_in; (void)out_size; (void)d_ws; (void)ws_size;
    const float* in   = (const float*)d_in[0];   // 4096*4096 fp32
    const float* wgt  = (const float*)d_in[1];   // 3*3 fp32
    const float* bias = (const float*)d_in[2];   // 1 fp32
    float* out        = (float*)d_out;           // 4094*4094 fp32

    dim3 grid((OUT_W + TILE - 1) / TILE, (OUT_W + TILE - 1) / TILE); // 64 x 64
    dim3 block(256);
    conv3x3_tdm_kernel<<<grid, block, LDS_FLOATS * sizeof(float), stream>>>(
        in, wgt, bias, out);
}